// SNS_Perception_42640435314879
// MI455X (gfx1250) — compile-verified
//
#include <hip/hip_runtime.h>
#include <cstdint>
#include <cstddef>

// ---------------------------------------------------------------------------
// SNS (liquid-time-constant) 3-cell step, fused.
//   - Trans/VALU bound => per (i,u): 1 fma + 1 v_tanh_f32 + 2 fma, with all
//     constant terms pre-folded (prep kernel -> packed d_ws buffer).
//   - Accumulation FMAs expressed as 2-wide vector fma to target
//     V_PK_FMA_F32 (CDNA5 VOP3P), halving VALU pressure vs scalar.
//   - Packed params staged to LDS via gfx1250 global_load_async_to_lds_b128
//     (ASYNCcnt) and consumed as ds_load_b128 broadcasts.
// ---------------------------------------------------------------------------

#define DEV __device__ __forceinline__

typedef float f32x2 __attribute__((ext_vector_type(2)));
typedef float f32x4 __attribute__((ext_vector_type(4)));

constexpr int   BATCH = 131072;
constexpr float DT    = 1.0f / 240.0f;

constexpr int S1_I = 9,  S1_U = 12;
constexpr int S2_I = 15, S2_U = 8;
constexpr int C_I  = 8,  C_U  = 8;

// Packed per-cell layout (floats):
//  SA[I*U] SB[I*U] SP[I*U] SW[I*U]  RA[U*U] RB[U*U] RP[U*U] RW[U*U]
//  IW[I] IB[I] DTT[U] NUMC[U] DENC[U] OW[U] OB[U]
constexpr int cell_size(int I, int U) { return 4*I*U + 4*U*U + 2*I + 5*U; }
constexpr int align4(int x) { return (x + 3) & ~3; }

constexpr int S1_BASE  = 0;
constexpr int S2_BASE  = S1_BASE + align4(cell_size(S1_I, S1_U));   // 1088
constexpr int CMD_BASE = S2_BASE + align4(cell_size(S2_I, S2_U));   // 1896
constexpr int NPACK    = CMD_BASE + align4(cell_size(C_I, C_U));    // 2464 floats
constexpr int NPACK16B = NPACK / 4;                                  // 616 x 16B

struct ParamPtrs { const float* p[42]; };

DEV float tanh_fast(float z) {
#if __has_builtin(__builtin_amdgcn_tanhf)
  return __builtin_amdgcn_tanhf(z);   // v_tanh_f32 (gfx1250 trans op)
#else
  return tanhf(z);
#endif
}

DEV f32x2 pk_fma(f32x2 a, f32x2 b, f32x2 c) {
  return __builtin_elementwise_fma(a, b, c);   // -> v_pk_fma_f32 if selected
}

// ------------------------------- prep kernel -------------------------------
DEV void fill_cell(int I, int U, float* c, const float* const* q) {
  const float* iw    = q[0];
  const float* ib    = q[1];
  const float* smu   = q[2];
  const float* ssig  = q[3];
  const float* serev = q[4];
  const float* sw    = q[5];
  const float* mu    = q[6];
  const float* sig   = q[7];
  const float* erev  = q[8];
  const float* w     = q[9];
  const float* tau   = q[10];
  const float* bb    = q[11];
  const float* ow    = q[12];
  const float* ob    = q[13];

  const int IU = I * U, UU = U * U;
  float* SA = c;          float* SB = SA + IU;  float* SP = SB + IU;  float* SW = SP + IU;
  float* RA = SW + IU;    float* RB = RA + UU;  float* RP = RB + UU;  float* RW = RP + UU;
  float* IWo = RW + UU;   float* IBo = IWo + I;
  float* DTT = IBo + I;   float* NUMC = DTT + U; float* DENC = NUMC + U;
  float* OW  = DENC + U;  float* OB = OW + U;

  for (int k = threadIdx.x; k < IU; k += blockDim.x) {
    float s = 0.5f * ssig[k];
    SA[k] = s;
    SB[k] = -smu[k] * s;
    float h = 0.5f * sw[k];
    SW[k] = h;
    SP[k] = h * serev[k];
  }
  for (int k = threadIdx.x; k < UU; k += blockDim.x) {
    float s = 0.5f * sig[k];
    RA[k] = s;
    RB[k] = -mu[k] * s;
    float h = 0.5f * w[k];
    RW[k] = h;
    RP[k] = h * erev[k];
  }
  for (int i = threadIdx.x; i < I; i += blockDim.x) { IWo[i] = iw[i]; IBo[i] = ib[i]; }
  for (int u = threadIdx.x; u < U; u += blockDim.x) {
    float sn = bb[u];      // fold b into NUMC
    float sd = 1.0f;       // fold the "-v" term's 1 into DENC
    for (int i = 0; i < I; i++) {
      float h = 0.5f * sw[i * U + u];
      sn += h * serev[i * U + u];
      sd += h;
    }
    for (int j = 0; j < U; j++) {
      float h = 0.5f * w[j * U + u];
      sn += h * erev[j * U + u];
      sd += h;
    }
    NUMC[u] = sn;
    DENC[u] = sd;
    DTT[u]  = DT / tau[u];
    OW[u]   = ow[u];
    OB[u]   = ob[u];
  }
}

__global__ void __launch_bounds__(256)
sns_prep_42640435314879(ParamPtrs P, float* __restrict__ ws) {
  fill_cell(S1_I, S1_U, ws + S1_BASE,  P.p + 0);
  fill_cell(S2_I, S2_U, ws + S2_BASE,  P.p + 14);
  fill_cell(C_I,  C_U,  ws + CMD_BASE, P.p + 28);
}

// ------------------------------- main kernel -------------------------------
// 4 (i,u) pairs: 4x ds_load_b128, 2 pk_fma (args), 4 v_tanh_f32,
// 4 pk_fma (num/den accumulate).
DEV void chunk4(float s, const float* A, const float* Bq, const float* Pq,
                const float* Wq, int off, f32x2* num, f32x2* den, int h) {
  f32x4 a = *(const f32x4*)(A + off);
  f32x4 b = *(const f32x4*)(Bq + off);
  f32x4 p = *(const f32x4*)(Pq + off);
  f32x4 w = *(const f32x4*)(Wq + off);
  f32x2 sv = {s, s};
  f32x2 g0 = pk_fma(sv, a.lo, b.lo);
  f32x2 g1 = pk_fma(sv, a.hi, b.hi);
  f32x2 t0 = {tanh_fast(g0.x), tanh_fast(g0.y)};
  f32x2 t1 = {tanh_fast(g1.x), tanh_fast(g1.y)};
  num[h + 0] = pk_fma(t0, p.lo, num[h + 0]);
  den[h + 0] = pk_fma(t0, w.lo, den[h + 0]);
  num[h + 1] = pk_fma(t1, p.hi, num[h + 1]);
  den[h + 1] = pk_fma(t1, w.hi, den[h + 1]);
}

template <int I, int U>
DEV void cell_step(const float* x, float* v, const float* sm, int base) {
  const int IU = I * U, UU = U * U;
  const float* SA = sm + base;   const float* SB = SA + IU;
  const float* SP = SB + IU;     const float* SW = SP + IU;
  const float* RA = SW + IU;     const float* RB = RA + UU;
  const float* RP = RB + UU;     const float* RW = RP + UU;
  const float* IWp = RW + UU;    const float* IBp = IWp + I;
  const float* DTT = IBp + I;    const float* NUMC = DTT + U;
  const float* DENC = NUMC + U;

  f32x2 num[U / 2], den[U / 2];
#pragma unroll
  for (int h = 0; h < U / 2; h++) {
    num[h] = *(const f32x2*)(NUMC + 2 * h);
    den[h] = *(const f32x2*)(DENC + 2 * h);
  }

  // sensory synapses
#pragma unroll
  for (int i = 0; i < I; i++) {
    float xm = fmaf(x[i], IWp[i], IBp[i]);
#pragma unroll
    for (int u = 0; u < U; u += 4)
      chunk4(xm, SA, SB, SP, SW, i * U + u, num, den, u / 2);
  }
  // recurrent synapses (use OLD state)
#pragma unroll
  for (int j = 0; j < U; j++) {
    float vj = v[j];
#pragma unroll
    for (int u = 0; u < U; u += 4)
      chunk4(vj, RA, RB, RP, RW, j * U + u, num, den, u / 2);
  }
  // v_new = v + dt/tau * (num - v*den)   (b, +1, const sums pre-folded)
#pragma unroll
  for (int h = 0; h < U / 2; h++) {
    f32x2 dtt = *(const f32x2*)(DTT + 2 * h);
    f32x2 vv  = {v[2 * h], v[2 * h + 1]};
    f32x2 r   = pk_fma(dtt, pk_fma(-vv, den[h], num[h]), vv);
    v[2 * h]     = r.x;
    v[2 * h + 1] = r.y;
  }
}

__global__ void __launch_bounds__(256)
sns_main_42640435314879(const float* __restrict__ gp, const float* __restrict__ op,
                        const float* __restrict__ tp, const float* __restrict__ fo,
                        const float* __restrict__ s1v, const float* __restrict__ s2v,
                        const float* __restrict__ cmdv, const float* __restrict__ pack,
                        float* __restrict__ out, float* __restrict__ force_out) {
  __shared__ __align__(16) float smem[NPACK];

  // ---- stage packed params to LDS via async-to-LDS (ASYNCcnt path) ----
  for (int k = threadIdx.x; k < NPACK16B; k += 256) {
    unsigned lds_off = (unsigned)(uintptr_t)(&smem[4 * k]);
    unsigned goff    = (unsigned)(k * 16);
    asm volatile("global_load_async_to_lds_b128 %0, %1, %2"
                 :
                 : "v"(lds_off), "v"(goff), "s"(pack)
                 : "memory");
  }
  asm volatile("s_wait_asynccnt 0" ::: "memory");
  __syncthreads();

  const int b = blockIdx.x * 256 + threadIdx.x;
  const size_t bs = (size_t)b;

  // ---- inputs ----
  float x1[S1_I];
#pragma unroll
  for (int k = 0; k < 3; k++) {
    x1[k]     = gp[bs * 3 + k];
    x1[3 + k] = op[bs * 3 + k];
    x1[6 + k] = tp[bs * 3 + k];
  }
  float v1[S1_U];
  {
    const float4* q = (const float4*)(s1v + bs * 12);
    float4 a = q[0], c = q[1], d = q[2];
    v1[0]=a.x; v1[1]=a.y; v1[2]=a.z; v1[3]=a.w;
    v1[4]=c.x; v1[5]=c.y; v1[6]=c.z; v1[7]=c.w;
    v1[8]=d.x; v1[9]=d.y; v1[10]=d.z; v1[11]=d.w;
  }
  float v2[S2_U];
  {
    const float4* q = (const float4*)(s2v + bs * 8);
    float4 a = q[0], c = q[1];
    v2[0]=a.x; v2[1]=a.y; v2[2]=a.z; v2[3]=a.w;
    v2[4]=c.x; v2[5]=c.y; v2[6]=c.z; v2[7]=c.w;
  }
  float v3[C_U];
  {
    const float4* q = (const float4*)(cmdv + bs * 8);
    float4 a = q[0], c = q[1];
    v3[0]=a.x; v3[1]=a.y; v3[2]=a.z; v3[3]=a.w;
    v3[4]=c.x; v3[5]=c.y; v3[6]=c.z; v3[7]=c.w;
  }

  // ---- cell 1: pos(9) -> s1(12) ----
  cell_step<S1_I, S1_U>(x1, v1, smem, S1_BASE);

  // ---- cell 2: concat(s1_new, force)(15) -> s2(8) ----
  float x2[S2_I];
#pragma unroll
  for (int u = 0; u < S1_U; u++) x2[u] = v1[u];
#pragma unroll
  for (int k = 0; k < 3; k++) x2[S1_U + k] = fo[bs * 3 + k];
  cell_step<S2_I, S2_U>(x2, v2, smem, S2_BASE);

  // ---- cell 3: s2_new(8) -> cmd(8) ----
  cell_step<C_I, C_U>(v2, v3, smem, CMD_BASE);

  // ---- outputs: out = v3*ow + ob ; force_out = s2_new[:, 7] ----
  const float* OW = smem + CMD_BASE + 4 * C_I * C_U + 4 * C_U * C_U + 2 * C_I + 3 * C_U;
  const float* OB = OW + C_U;
  float4 o0, o1;
  o0.x = fmaf(v3[0], OW[0], OB[0]); o0.y = fmaf(v3[1], OW[1], OB[1]);
  o0.z = fmaf(v3[2], OW[2], OB[2]); o0.w = fmaf(v3[3], OW[3], OB[3]);
  o1.x = fmaf(v3[4], OW[4], OB[4]); o1.y = fmaf(v3[5], OW[5], OB[5]);
  o1.z = fmaf(v3[6], OW[6], OB[6]); o1.w = fmaf(v3[7], OW[7], OB[7]);
  float4* oq = (float4*)(out + bs * 8);
  oq[0] = o0;
  oq[1] = o1;
  force_out[bs] = v2[7];
}

// ------------------------------- launch ------------------------------------
extern "C" void kernel_launch(void* const* d_in, const int* in_sizes, int n_in,
                              void* d_out, int out_size, void* d_ws, size_t ws_size,
                              hipStream_t stream) {
  (void)in_sizes; (void)n_in; (void)out_size; (void)ws_size;

  ParamPtrs P;
  for (int i = 0; i < 42; i++) P.p[i] = (const float*)d_in[7 + i];
  float* ws = (float*)d_ws;

  sns_prep_42640435314879<<<1, 256, 0, stream>>>(P, ws);

  const float* gp   = (const float*)d_in[0];
  const float* op   = (const float*)d_in[1];
  const float* tp   = (const float*)d_in[2];
  const float* fo   = (const float*)d_in[3];
  const float* s1v  = (const float*)d_in[4];
  const float* s2v  = (const float*)d_in[5];
  const float* cmdv = (const float*)d_in[6];
  float* out = (float*)d_out;
  float* force_out = out + (size_t)BATCH * 8;

  sns_main_42640435314879<<<BATCH / 256, 256, 0, stream>>>(
      gp, op, tp, fo, s1v, s2v, cmdv, ws, out, force_out);
}